// GATConv_71657234366695
// MI455X (gfx1250) — compile-verified
//
#include <hip/hip_runtime.h>
#include <math.h>

// Problem constants (match reference)
#define NNODE 8192
#define DDIM  256
#define HHEAD 2
#define RCHAN 16
#define NEDGE 32768
#define NEF   131072

typedef __attribute__((ext_vector_type(2))) float v2f;
typedef __attribute__((ext_vector_type(8))) float v8f;

__device__ __forceinline__ float silu_f(float v) { return v / (1.f + __expf(-v)); }

__device__ __forceinline__ float wave_sum(float v) {
#pragma unroll
  for (int off = 16; off > 0; off >>= 1) v += __shfl_xor(v, off, 32);
  return v;
}

// fp32 WMMA: D = A(16x4) * B(4x16) + C, f32 accumulate (CDNA5 V_WMMA_F32_16X16X4_F32)
__device__ __forceinline__ v8f wmma16x16x4(v2f a, v2f b, v8f c) {
#if __has_builtin(__builtin_amdgcn_wmma_f32_16x16x4_f32)
  return __builtin_amdgcn_wmma_f32_16x16x4_f32(false, a, false, b, (short)0, c,
                                               false, false);
#else
  return c; // fallback so compile never fails; histogram would show wmma==0
#endif
}

// ---------------------------------------------------------------------------
// Fused triple LayerNorm: one read of x, three scaled outputs.
// One wave per row (wave32; lane handles 8 of 256 columns).
// ---------------------------------------------------------------------------
__global__ void ln3_kernel(const float* __restrict__ x,
                           const float* __restrict__ g1, const float* __restrict__ be1,
                           const float* __restrict__ g2, const float* __restrict__ be2,
                           const float* __restrict__ g3, const float* __restrict__ be3,
                           float* __restrict__ xn1, float* __restrict__ xn2,
                           float* __restrict__ xn3) {
  int row  = blockIdx.x * (blockDim.x >> 5) + (threadIdx.x >> 5);
  int lane = threadIdx.x & 31;
  if (row >= NNODE) return;
  size_t base = (size_t)row * DDIM;
  float xv[8], s = 0.f, sq = 0.f;
#pragma unroll
  for (int i = 0; i < 8; ++i) {
    xv[i] = x[base + lane + 32 * i];
    s += xv[i];
    sq += xv[i] * xv[i];
  }
  s = wave_sum(s);
  sq = wave_sum(sq);
  float mean = s * (1.f / DDIM);
  float var  = sq * (1.f / DDIM) - mean * mean;
  float rstd = rsqrtf(var + 1e-5f);
#pragma unroll
  for (int i = 0; i < 8; ++i) {
    int c = lane + 32 * i;
    float nv = (xv[i] - mean) * rstd;
    xn1[base + c] = nv * g1[c] + be1[c];
    xn2[base + c] = nv * g2[c] + be2[c];
    xn3[base + c] = nv * g3[c] + be3[c];
  }
}

// ---------------------------------------------------------------------------
// fp32 WMMA GEMM: C[M,N] = act(A[M,K] @ B[K,N] + bias) (+ resid)
// One wave computes a 16 x (16*NT) strip: per K-step one A-frag (v2f, 8B
// contiguous load) is reused across NT chained V_WMMA_F32_16X16X4_F32 with
// independent accumulators (no D->A/B RAW hazard => no NOP insertion).
// ---------------------------------------------------------------------------
template <int ACT, int NT> // ACT: 0 = identity, 1 = SiLU
__global__ __launch_bounds__(32)
void gemm_wmma_kernel(const float* __restrict__ A, const float* __restrict__ B,
                      const float* __restrict__ bias,
                      const float* __restrict__ resid, float* __restrict__ C,
                      int M, int Nn, int K) {
  int lane = threadIdx.x & 31;
  int lr   = lane & 15;
  int half = lane >> 4;
  int tn = blockIdx.x * (16 * NT);
  int tm = blockIdx.y * 16;
  const float* Arow = A + (size_t)(tm + lr) * K;
  v8f acc[NT];
#pragma unroll
  for (int t = 0; t < NT; ++t) acc[t] = {};
  for (int k = 0; k < K; k += 4) {
    int ka = k + half * 2;
    v2f a = *(const v2f*)(Arow + ka); // 8B-aligned (ka even)
    const float* B0 = B + (size_t)ka * Nn + tn + lr;
    const float* B1 = B + (size_t)(ka + 1) * Nn + tn + lr;
#pragma unroll
    for (int t = 0; t < NT; ++t) {
      v2f b;
      b[0] = B0[t * 16];
      b[1] = B1[t * 16];
      acc[t] = wmma16x16x4(a, b, acc[t]);
    }
  }
  int rowbase = tm + half * 8;
#pragma unroll
  for (int t = 0; t < NT; ++t) {
    int col = tn + t * 16 + lr;
    float bv = bias[col];
#pragma unroll
    for (int v = 0; v < 8; ++v) {
      float val = acc[t][v] + bv;
      if (ACT == 1) val = silu_f(val);
      if (resid) val += resid[(size_t)(rowbase + v) * Nn + col];
      C[(size_t)(rowbase + v) * Nn + col] = val;
    }
  }
}

// ---------------------------------------------------------------------------
// av1 = xn3@Wa1+ba1, av2 = xn3@Wa2+ba2  (N x 2 each) — one wave per node.
// ---------------------------------------------------------------------------
__global__ void attvec_kernel(const float* __restrict__ xn3,
                              const float* __restrict__ Wa1, const float* __restrict__ ba1,
                              const float* __restrict__ Wa2, const float* __restrict__ ba2,
                              float* __restrict__ av1, float* __restrict__ av2) {
  int node = blockIdx.x * (blockDim.x >> 5) + (threadIdx.x >> 5);
  int lane = threadIdx.x & 31;
  if (node >= NNODE) return;
  const float* xr = xn3 + (size_t)node * DDIM;
  float a10 = 0, a11 = 0, a20 = 0, a21 = 0;
#pragma unroll
  for (int i = 0; i < 8; ++i) {
    int k = lane + 32 * i;
    float v = xr[k];
    a10 += v * Wa1[k * 2 + 0];
    a11 += v * Wa1[k * 2 + 1];
    a20 += v * Wa2[k * 2 + 0];
    a21 += v * Wa2[k * 2 + 1];
  }
  a10 = wave_sum(a10);
  a11 = wave_sum(a11);
  a20 = wave_sum(a20);
  a21 = wave_sum(a21);
  if (lane == 0) {
    av1[node * 2 + 0] = a10 + ba1[0];
    av1[node * 2 + 1] = a11 + ba1[1];
    av2[node * 2 + 0] = a20 + ba2[0];
    av2[node * 2 + 1] = a21 + ba2[1];
  }
}

// Zero accumulators + set segment-max to -inf-ish. Re-run every call.
__global__ void init_kernel(float* __restrict__ nxloc, float* __restrict__ nxfc,
                            float* __restrict__ sseg, float* __restrict__ mseg,
                            int nd, int nh) {
  int i = blockIdx.x * blockDim.x + threadIdx.x;
  int stride = gridDim.x * blockDim.x;
  for (int j = i; j < nd; j += stride) {
    nxloc[j] = 0.f;
    nxfc[j] = 0.f;
  }
  for (int j = i; j < nh; j += stride) {
    sseg[j] = 0.f;
    mseg[j] = -3.0e38f;
  }
}

// ---------------------------------------------------------------------------
// Local message passing (E real edges + N self-loops), one wave per edge.
// msg = mean_h silu(xs[src,h]+xt[dst,h]+eemb[h]) scattered via f32 atomics.
// ---------------------------------------------------------------------------
__global__ void edge_local_kernel(const float* __restrict__ xs, const float* __restrict__ xt,
                                  const int* __restrict__ ei, const int* __restrict__ eattr,
                                  const float* __restrict__ E1, const float* __restrict__ E2,
                                  const float* __restrict__ E3, float* __restrict__ nxloc) {
  int e = blockIdx.x * (blockDim.x >> 5) + (threadIdx.x >> 5);
  int lane = threadIdx.x & 31;
  if (e >= NEDGE + NNODE) return;
  int src, dst, a0, a1, a2;
  if (e < NEDGE) {
    src = ei[e];
    dst = ei[NEDGE + e];
    a0 = eattr[e * 3 + 0];
    a1 = eattr[e * 3 + 1];
    a2 = eattr[e * 3 + 2];
  } else { // self loop, attr = (25,0,0)
    src = dst = e - NEDGE;
    a0 = 25; a1 = 0; a2 = 0;
  }
  const float* xs0 = xs + (size_t)src * (HHEAD * DDIM);
  const float* xt0 = xt + (size_t)dst * (HHEAD * DDIM);
  const float* e1 = E1 + (size_t)a0 * (HHEAD * DDIM);
  const float* e2 = E2 + (size_t)a1 * (HHEAD * DDIM);
  const float* e3 = E3 + (size_t)a2 * (HHEAD * DDIM);
  float* outp = nxloc + (size_t)dst * DDIM;
#pragma unroll
  for (int i = 0; i < 8; ++i) {
    int d = lane + 32 * i;
    float m0 = xs0[d] + xt0[d] + e1[d] + e2[d] + e3[d];
    float m1 = xs0[DDIM + d] + xt0[DDIM + d] + e1[DDIM + d] + e2[DDIM + d] + e3[DDIM + d];
    atomicAdd(outp + d, 0.5f * (silu_f(m0) + silu_f(m1)));
  }
}

// ---------------------------------------------------------------------------
// Segment softmax over fc edges (3 passes; logits buffer reused for exp vals)
// ---------------------------------------------------------------------------
__device__ __forceinline__ void atomic_max_float(float* addr, float val) {
  if (val >= 0.f)
    atomicMax((int*)addr, __float_as_int(val));
  else
    atomicMin((unsigned int*)addr, __float_as_uint(val));
}

__global__ void fc_logits_kernel(const int* __restrict__ eia,
                                 const float* __restrict__ av1, const float* __restrict__ av2,
                                 const float* __restrict__ Wfc, const float* __restrict__ bfc,
                                 float* __restrict__ logits, float* __restrict__ mseg) {
  int e = blockIdx.x * blockDim.x + threadIdx.x;
  if (e >= NEF) return;
  int src = eia[e], dst = eia[NEF + e];
  float a0 = av1[src * 2 + 0] + av2[dst * 2 + 0];
  float a1 = av1[src * 2 + 1] + av2[dst * 2 + 1];
  a0 = a0 > 0.f ? a0 : 0.2f * a0; // leaky_relu 0.2
  a1 = a1 > 0.f ? a1 : 0.2f * a1;
  float l0 = a0 * Wfc[0] + a1 * Wfc[2] + bfc[0];
  float l1 = a0 * Wfc[1] + a1 * Wfc[3] + bfc[1];
  logits[e * 2 + 0] = l0;
  logits[e * 2 + 1] = l1;
  atomic_max_float(&mseg[dst * 2 + 0], l0);
  atomic_max_float(&mseg[dst * 2 + 1], l1);
}

__global__ void fc_exp_kernel(const int* __restrict__ eia, float* __restrict__ logits,
                              const float* __restrict__ mseg, float* __restrict__ sseg) {
  int e = blockIdx.x * blockDim.x + threadIdx.x;
  if (e >= NEF) return;
  int dst = eia[NEF + e];
  float e0 = __expf(logits[e * 2 + 0] - mseg[dst * 2 + 0]);
  float e1 = __expf(logits[e * 2 + 1] - mseg[dst * 2 + 1]);
  logits[e * 2 + 0] = e0;
  logits[e * 2 + 1] = e1;
  atomicAdd(&sseg[dst * 2 + 0], e0);
  atomicAdd(&sseg[dst * 2 + 1], e1);
}

// ---------------------------------------------------------------------------
// fc scatter GEMM: batch 16 edges per wave.
// U[16 edges x 16 rc] = 0.5 * sum_h attn_h * silu(s3[src,h]+s4[dst,h])
// out[16x256] = U @ W5 (+ b5 * attn-sum), atomic-scattered by dst.
// K=16 via 4 chained V_WMMA_F32_16X16X4_F32; A-frag staged in LDS.
// ---------------------------------------------------------------------------
__global__ __launch_bounds__(32)
void fc_scatter_wmma_kernel(const int* __restrict__ eia,
                            const float* __restrict__ s3, const float* __restrict__ s4,
                            const float* __restrict__ expv, const float* __restrict__ sseg,
                            const float* __restrict__ W5, const float* __restrict__ b5,
                            float* __restrict__ nxfc) {
  __shared__ float Ulds[16 * 16];
  __shared__ float wsum[16];
  __shared__ int dstl[16];
  int lane = threadIdx.x;
  int el = lane & 15, half = lane >> 4;
  int e = blockIdx.x * 16 + el;
  int src = eia[e], dst = eia[NEF + e];
  float at0 = expv[e * 2 + 0] / (sseg[dst * 2 + 0] + 1e-16f);
  float at1 = expv[e * 2 + 1] / (sseg[dst * 2 + 1] + 1e-16f);
  const float* s3p = s3 + (size_t)src * (HHEAD * RCHAN);
  const float* s4p = s4 + (size_t)dst * (HHEAD * RCHAN);
#pragma unroll
  for (int i = 0; i < 8; ++i) {
    int r = half * 8 + i;
    float u0 = s3p[r] + s4p[r];
    float u1 = s3p[RCHAN + r] + s4p[RCHAN + r];
    Ulds[el * 16 + r] = 0.5f * (at0 * silu_f(u0) + at1 * silu_f(u1));
  }
  if (half == 0) {
    wsum[el] = 0.5f * (at0 + at1);
    dstl[el] = dst;
  }
  __syncthreads();
  int lr = el;
  v2f aF[4];
#pragma unroll
  for (int kk = 0; kk < 4; ++kk) {
    int ka = kk * 4 + half * 2;
    aF[kk][0] = Ulds[lr * 16 + ka];
    aF[kk][1] = Ulds[lr * 16 + ka + 1];
  }
  float bw[8];
  int dl[8];
#pragma unroll
  for (int v = 0; v < 8; ++v) {
    int r = half * 8 + v;
    bw[v] = wsum[r];
    dl[v] = dstl[r];
  }
  for (int nt = 0; nt < 16; ++nt) {
    int col = nt * 16 + lr;
    v8f acc = {};
#pragma unroll
    for (int kk = 0; kk < 4; ++kk) {
      int ka = kk * 4 + half * 2;
      v2f b;
      b[0] = W5[ka * DDIM + col];
      b[1] = W5[(ka + 1) * DDIM + col];
      acc = wmma16x16x4(aF[kk], b, acc);
    }
    float bv = b5[col];
#pragma unroll
    for (int v = 0; v < 8; ++v)
      atomicAdd(&nxfc[(size_t)dl[v] * DDIM + col], acc[v] + bv * bw[v]);
  }
}

// ---------------------------------------------------------------------------
// new_x = nxloc + 0.01*nxfc + x; y = LN(new_x, g4, be4). One wave per row.
// ---------------------------------------------------------------------------
__global__ void combine_ln_kernel(const float* __restrict__ x, const float* __restrict__ nxloc,
                                  const float* __restrict__ nxfc,
                                  const float* __restrict__ g4, const float* __restrict__ be4,
                                  float* __restrict__ newx, float* __restrict__ y) {
  int row = blockIdx.x * (blockDim.x >> 5) + (threadIdx.x >> 5);
  int lane = threadIdx.x & 31;
  if (row >= NNODE) return;
  size_t base = (size_t)row * DDIM;
  float xv[8], s = 0.f, sq = 0.f;
#pragma unroll
  for (int i = 0; i < 8; ++i) {
    int c = lane + 32 * i;
    float v = nxloc[base + c] + 0.01f * nxfc[base + c] + x[base + c];
    newx[base + c] = v;
    xv[i] = v;
    s += v;
    sq += v * v;
  }
  s = wave_sum(s);
  sq = wave_sum(sq);
  float mean = s * (1.f / DDIM);
  float var  = sq * (1.f / DDIM) - mean * mean;
  float rstd = rsqrtf(var + 1e-5f);
#pragma unroll
  for (int i = 0; i < 8; ++i) {
    int c = lane + 32 * i;
    y[base + c] = (xv[i] - mean) * rstd * g4[c] + be4[c];
  }
}

// ---------------------------------------------------------------------------
extern "C" void kernel_launch(void* const* d_in, const int* in_sizes, int n_in,
                              void* d_out, int out_size, void* d_ws, size_t ws_size,
                              hipStream_t stream) {
  (void)in_sizes; (void)n_in; (void)out_size; (void)ws_size;
  const float* x    = (const float*)d_in[0];
  const int* ei     = (const int*)d_in[1];
  const int* eia    = (const int*)d_in[2];
  const int* eattr  = (const int*)d_in[3];
  const float* W1   = (const float*)d_in[4];  const float* b1  = (const float*)d_in[5];
  const float* W2   = (const float*)d_in[6];  const float* b2  = (const float*)d_in[7];
  const float* W3   = (const float*)d_in[8];  const float* b3  = (const float*)d_in[9];
  const float* W4   = (const float*)d_in[10]; const float* b4  = (const float*)d_in[11];
  const float* W5   = (const float*)d_in[12]; const float* b5  = (const float*)d_in[13];
  const float* Wa1  = (const float*)d_in[14]; const float* ba1 = (const float*)d_in[15];
  const float* Wa2  = (const float*)d_in[16]; const float* ba2 = (const float*)d_in[17];
  const float* Wfc  = (const float*)d_in[18]; const float* bfc = (const float*)d_in[19];
  const float* E1   = (const float*)d_in[20];
  const float* E2   = (const float*)d_in[21];
  const float* E3   = (const float*)d_in[22];
  const float* g1   = (const float*)d_in[23]; const float* be1 = (const float*)d_in[24];
  const float* g2   = (const float*)d_in[25]; const float* be2 = (const float*)d_in[26];
  const float* g3   = (const float*)d_in[27]; const float* be3 = (const float*)d_in[28];
  const float* g4   = (const float*)d_in[29]; const float* be4 = (const float*)d_in[30];
  const float* Wf1  = (const float*)d_in[31]; const float* bf1 = (const float*)d_in[32];
  const float* Wf2  = (const float*)d_in[33]; const float* bf2 = (const float*)d_in[34];
  float* out = (float*)d_out;

  // Workspace carve-up (floats). Total ~31M floats (~125 MB).
  float* ws = (float*)d_ws;
  size_t off = 0;
  auto carve = [&](size_t n) { float* p = ws + off; off += n; return p; };
  const size_t ND = (size_t)NNODE * DDIM;
  float* xn1   = carve(ND);
  float* xn2   = carve(ND);
  float* xn3   = carve(ND);
  float* xs    = carve(ND * HHEAD);
  float* xt    = carve(ND * HHEAD);
  float* s3b   = carve((size_t)NNODE * HHEAD * RCHAN);
  float* s4b   = carve((size_t)NNODE * HHEAD * RCHAN);
  float* av1   = carve((size_t)NNODE * HHEAD);
  float* av2   = carve((size_t)NNODE * HHEAD);
  float* logb  = carve((size_t)NEF * HHEAD); // logits, reused as exp values
  float* mseg  = carve((size_t)NNODE * HHEAD);
  float* sseg  = carve((size_t)NNODE * HHEAD);
  float* nxloc = carve(ND);
  float* nxfc  = carve(ND);
  float* newx  = carve(ND);
  float* yb    = carve(ND);
  float* tb    = carve(ND * 4);

  // 1) fused triple LayerNorm
  ln3_kernel<<<NNODE / 8, 256, 0, stream>>>(x, g1, be1, g2, be2, g3, be3, xn1, xn2, xn3);

  // 2) dense WMMA GEMMs for source/target transforms (16x64 strip per wave)
  dim3 gxs(512 / 64, NNODE / 16);
  gemm_wmma_kernel<0, 4><<<gxs, 32, 0, stream>>>(xn1, W1, b1, nullptr, xs, NNODE, 512, DDIM);
  gemm_wmma_kernel<0, 4><<<gxs, 32, 0, stream>>>(xn1, W2, b2, nullptr, xt, NNODE, 512, DDIM);
  dim3 gs(32 / 32, NNODE / 16);
  gemm_wmma_kernel<0, 2><<<gs, 32, 0, stream>>>(xn2, W3, b3, nullptr, s3b, NNODE, 32, DDIM);
  gemm_wmma_kernel<0, 2><<<gs, 32, 0, stream>>>(xn2, W4, b4, nullptr, s4b, NNODE, 32, DDIM);

  // 3) attention projection vectors (tiny GEMV, wave-reduced)
  attvec_kernel<<<NNODE / 8, 256, 0, stream>>>(xn3, Wa1, ba1, Wa2, ba2, av1, av2);

  // 4) zero accumulators / init segment-max (deterministic every call)
  init_kernel<<<1024, 256, 0, stream>>>(nxloc, nxfc, sseg, mseg, (int)ND,
                                        NNODE * HHEAD);

  // 5) local message passing (E edges + N self loops)
  edge_local_kernel<<<(NEDGE + NNODE) / 8, 256, 0, stream>>>(xs, xt, ei, eattr, E1, E2,
                                                             E3, nxloc);

  // 6) segment softmax over fc edges
  fc_logits_kernel<<<NEF / 256, 256, 0, stream>>>(eia, av1, av2, Wfc, bfc, logb, mseg);
  fc_exp_kernel<<<NEF / 256, 256, 0, stream>>>(eia, logb, mseg, sseg);

  // 7) fc message GEMM (16 edges/wave, WMMA vs W5) + atomic scatter
  fc_scatter_wmma_kernel<<<NEF / 16, 32, 0, stream>>>(eia, s3b, s4b, logb, sseg, W5, b5,
                                                      nxfc);

  // 8) residual combine + LN4
  combine_ln_kernel<<<NNODE / 8, 256, 0, stream>>>(x, nxloc, nxfc, g4, be4, newx, yb);

  // 9) FFN: t = silu(y@Wf1+bf1); out = t@Wf2 + bf2 + new_x
  dim3 gt(1024 / 64, NNODE / 16);
  gemm_wmma_kernel<1, 4><<<gt, 32, 0, stream>>>(yb, Wf1, bf1, nullptr, tb, NNODE, 1024, DDIM);
  dim3 go(DDIM / 64, NNODE / 16);
  gemm_wmma_kernel<0, 4><<<go, 32, 0, stream>>>(tb, Wf2, bf2, newx, out, NNODE, DDIM, 1024);
}